// GCN_5342939316758
// MI455X (gfx1250) — compile-verified
//
#include <hip/hip_runtime.h>
#include <hip/hip_bf16.h>

typedef __attribute__((ext_vector_type(2))) float v2f_t;
typedef __attribute__((ext_vector_type(8))) float v8f_t;

#define N_NODES 50000
#define N_EDGES 640000
#define IN_CH   128
#define HID_CH  128
#define OUT_CH  40
#define OUT_PAD 48   // pad 40 -> 48 so N-tiles stay multiples of 16

// ---------------- degree / normalization ----------------

__global__ void k_deg_init(float* __restrict__ deg) {
    int i = blockIdx.x * blockDim.x + threadIdx.x;
    if (i < N_NODES) deg[i] = 1.0f;              // self-loop contributes 1
}

__global__ void k_deg_count(const long long* __restrict__ dst, float* __restrict__ deg) {
    int e = blockIdx.x * blockDim.x + threadIdx.x;
    if (e < N_EDGES) atomicAdd(&deg[(int)dst[e]], 1.0f);
}

__global__ void k_dinv(const float* __restrict__ deg, float* __restrict__ dinv) {
    int i = blockIdx.x * blockDim.x + threadIdx.x;
    if (i < N_NODES) {
        float d = deg[i];
        dinv[i] = d > 0.0f ? rsqrtf(d) : 0.0f;
    }
}

__global__ void k_pad_w2(const float* __restrict__ W2, float* __restrict__ W2p) {
    int i = blockIdx.x * blockDim.x + threadIdx.x;
    if (i < HID_CH * OUT_PAD) {
        int k = i / OUT_PAD;
        int n = i - k * OUT_PAD;
        W2p[i] = (n < OUT_CH) ? W2[k * OUT_CH + n] : 0.0f;
    }
}

// ---------------- fp32 WMMA GEMM: C[MxNc] = A[MxK] * B[KxNc], row major ----
// Requires M%16==0, K%4==0, Nc%16==0. One wave computes one 16x16 C tile via
// K/4 chained v_wmma_f32_16x16x4_f32 ops.
//
// A-frag (16x4 f32): lanes 0-15 -> M=lane, K={0,1} in v[0],v[1];
//                    lanes 16-31 -> M=lane-16, K={2,3}.
// B-frag (4x16 f32): lanes 0-15 -> N=lane, K={0,1}; lanes 16-31 -> N=lane-16, K={2,3}.
// C/D (16x16 f32, 8 VGPRs): VGPR r -> M=r (lanes 0-15) / M=r+8 (lanes 16-31), N=lane&15.

__global__ void k_gemm_wmma_f32(const float* __restrict__ A, const float* __restrict__ B,
                                float* __restrict__ C, int M, int K, int Nc) {
    int wave = blockIdx.x * (blockDim.x >> 5) + (threadIdx.x >> 5);
    int lane = threadIdx.x & 31;
    int tilesN = Nc >> 4;
    int tiles  = (M >> 4) * tilesN;
    if (wave >= tiles) return;                   // wave-uniform exit (EXEC stays all-1)

    int tm = wave / tilesN;
    int tn = wave - tm * tilesN;

    int mrow = lane & 15;
    int ncol = lane & 15;
    int koff = (lane >> 4) << 1;                 // 0 for lanes 0-15, 2 for lanes 16-31

    const float* Arow = A + (size_t)(tm * 16 + mrow) * K + koff;
    const float* Bcol = B + (size_t)koff * Nc + tn * 16 + ncol;

    v8f_t acc = {};
    for (int k = 0; k < K; k += 4) {
        v2f_t a = *(const v2f_t*)(Arow + k);     // A[m][k+koff .. k+koff+1], 8B aligned
        v2f_t b;
        b.x = Bcol[(size_t)k * Nc];              // B[k+koff  ][n]
        b.y = Bcol[(size_t)(k + 1) * Nc];        // B[k+koff+1][n]
        acc = __builtin_amdgcn_wmma_f32_16x16x4_f32(
            false, a, false, b, (short)0, acc, false, false);
    }

    int rbase = (lane >> 4) << 3;                // lanes>=16 hold rows M=r+8
    float* Crow = C + (size_t)(tm * 16 + rbase) * Nc + tn * 16 + ncol;
#pragma unroll
    for (int r = 0; r < 8; ++r)
        Crow[(size_t)r * Nc] = acc[r];
}

// ---------------- layer-1 aggregation ----------------

// agg[i][c] = xw[i][c] * dinv[i]^2   (self-loop message; also zero-initializes agg)
__global__ void k_selfmsg1(const float* __restrict__ xw, const float* __restrict__ dinv,
                           float* __restrict__ agg) {
    int idx = blockIdx.x * blockDim.x + threadIdx.x;
    if (idx >= N_NODES * 32) return;
    int i  = idx >> 5;
    int cg = (idx & 31) << 2;
    float n = dinv[i] * dinv[i];
    const float4 v = *(const float4*)(xw + (size_t)i * HID_CH + cg);
    float4 r; r.x = v.x * n; r.y = v.y * n; r.z = v.z * n; r.w = v.w * n;
    *(float4*)(agg + (size_t)i * HID_CH + cg) = r;
}

// per-edge scatter: agg[dst] += xw[src] * dinv[src]*dinv[dst]   (128 ch = 32 float4 groups)
__global__ void k_scatter1(const long long* __restrict__ src, const long long* __restrict__ dst,
                           const float* __restrict__ dinv, const float* __restrict__ xw,
                           float* __restrict__ agg) {
    int idx = blockIdx.x * blockDim.x + threadIdx.x;
    if (idx >= N_EDGES * 32) return;
    int e  = idx >> 5;
    int cg = (idx & 31) << 2;
    int s = (int)src[e];
    int d = (int)dst[e];
    float norm = dinv[s] * dinv[d];
    const float4 v = *(const float4*)(xw + (size_t)s * HID_CH + cg);
    float* o = agg + (size_t)d * HID_CH + cg;
    atomicAdd(o + 0, v.x * norm);
    atomicAdd(o + 1, v.y * norm);
    atomicAdd(o + 2, v.z * norm);
    atomicAdd(o + 3, v.w * norm);
}

// h = relu(agg + b1), in place
__global__ void k_bias_relu(float* __restrict__ agg, const float* __restrict__ b1) {
    int idx = blockIdx.x * blockDim.x + threadIdx.x;
    if (idx >= N_NODES * 32) return;
    int i  = idx >> 5;
    int cg = (idx & 31) << 2;
    float4 v = *(const float4*)(agg + (size_t)i * HID_CH + cg);
    v.x = fmaxf(v.x + b1[cg + 0], 0.0f);
    v.y = fmaxf(v.y + b1[cg + 1], 0.0f);
    v.z = fmaxf(v.z + b1[cg + 2], 0.0f);
    v.w = fmaxf(v.w + b1[cg + 3], 0.0f);
    *(float4*)(agg + (size_t)i * HID_CH + cg) = v;
}

// ---------------- layer-2 aggregation ----------------

// out[i][c] = hw[i][c] * dinv[i]^2 + b2[c]   (self-loop + bias; zero-initializes out)
__global__ void k_out_init(const float* __restrict__ hw, const float* __restrict__ dinv,
                           const float* __restrict__ b2, float* __restrict__ out) {
    int idx = blockIdx.x * blockDim.x + threadIdx.x;
    if (idx >= N_NODES * 10) return;
    int i  = idx / 10;
    int cg = (idx - i * 10) << 2;
    float n = dinv[i] * dinv[i];
    const float4 v = *(const float4*)(hw + (size_t)i * OUT_PAD + cg);
    float4 r;
    r.x = v.x * n + b2[cg + 0];
    r.y = v.y * n + b2[cg + 1];
    r.z = v.z * n + b2[cg + 2];
    r.w = v.w * n + b2[cg + 3];
    *(float4*)(out + (size_t)i * OUT_CH + cg) = r;   // 40*4B=160B row stride, 16B aligned
}

__global__ void k_scatter2(const long long* __restrict__ src, const long long* __restrict__ dst,
                           const float* __restrict__ dinv, const float* __restrict__ hw,
                           float* __restrict__ out) {
    int idx = blockIdx.x * blockDim.x + threadIdx.x;
    if (idx >= N_EDGES * 10) return;
    int e  = idx / 10;
    int cg = (idx - e * 10) << 2;
    int s = (int)src[e];
    int d = (int)dst[e];
    float norm = dinv[s] * dinv[d];
    const float4 v = *(const float4*)(hw + (size_t)s * OUT_PAD + cg);
    float* o = out + (size_t)d * OUT_CH + cg;
    atomicAdd(o + 0, v.x * norm);
    atomicAdd(o + 1, v.y * norm);
    atomicAdd(o + 2, v.z * norm);
    atomicAdd(o + 3, v.w * norm);
}

// ---------------- launch ----------------

extern "C" void kernel_launch(void* const* d_in, const int* in_sizes, int n_in,
                              void* d_out, int out_size, void* d_ws, size_t ws_size,
                              hipStream_t stream) {
    (void)in_sizes; (void)n_in; (void)out_size; (void)ws_size;

    const float*     x   = (const float*)d_in[0];
    const long long* ei  = (const long long*)d_in[1];   // int64 [2, E]
    const float*     W1  = (const float*)d_in[2];
    const float*     b1  = (const float*)d_in[3];
    const float*     W2  = (const float*)d_in[4];
    const float*     b2  = (const float*)d_in[5];
    float*           out = (float*)d_out;

    const long long* src = ei;
    const long long* dst = ei + N_EDGES;

    // workspace layout (256B aligned offsets)
    char* ws = (char*)d_ws;
    float* deg  = (float*)(ws + 0);                         //  200,000 B
    float* dinv = (float*)(ws + (size_t)256 * 1024);        //  200,000 B
    float* W2p  = (float*)(ws + (size_t)512 * 1024);        //   24,576 B
    float* xw   = (float*)(ws + (size_t)1048576);           // 25.6 MB
    float* agg  = (float*)(ws + (size_t)26648576);          // 25.6 MB (becomes h)
    float* hw   = (float*)(ws + (size_t)52248576);          //  9.6 MB (padded, stride 48)

    const int T = 256;

    k_deg_init <<<(N_NODES + T - 1) / T, T, 0, stream>>>(deg);
    k_deg_count<<<(N_EDGES + T - 1) / T, T, 0, stream>>>(dst, deg);
    k_dinv     <<<(N_NODES + T - 1) / T, T, 0, stream>>>(deg, dinv);
    k_pad_w2   <<<(HID_CH * OUT_PAD + T - 1) / T, T, 0, stream>>>(W2, W2p);

    // GEMM1: xw[50000x128] = x @ W1 ; tiles = 3125*8 = 25000 waves, 8 waves/block
    {
        int tiles  = (N_NODES / 16) * (HID_CH / 16);
        int blocks = (tiles + 7) / 8;
        k_gemm_wmma_f32<<<blocks, T, 0, stream>>>(x, W1, xw, N_NODES, IN_CH, HID_CH);
    }

    k_selfmsg1 <<<(N_NODES * 32 + T - 1) / T, T, 0, stream>>>(xw, dinv, agg);
    k_scatter1 <<<(N_EDGES * 32 + T - 1) / T, T, 0, stream>>>(src, dst, dinv, xw, agg);
    k_bias_relu<<<(N_NODES * 32 + T - 1) / T, T, 0, stream>>>(agg, b1);

    // GEMM2: hw[50000x48] = h @ W2p ; tiles = 3125*3 = 9375 waves
    {
        int tiles  = (N_NODES / 16) * (OUT_PAD / 16);
        int blocks = (tiles + 7) / 8;
        k_gemm_wmma_f32<<<blocks, T, 0, stream>>>(agg, W2p, hw, N_NODES, HID_CH, OUT_PAD);
    }

    k_out_init <<<(N_NODES * 10 + T - 1) / T, T, 0, stream>>>(hw, dinv, b2, out);
    k_scatter2 <<<(N_EDGES * 10 + T - 1) / T, T, 0, stream>>>(src, dst, dinv, hw, out);
}